// chamfer_2DDist_5669356836186
// MI455X (gfx1250) — compile-verified
//
#include <hip/hip_runtime.h>

// CDNA5 / gfx1250 chamfer distance via V_WMMA_F32_16X16X4_F32.
//
// d(q,r) = |q|^2 + |r|^2 - 2 q.r  ==  K=4 dot of augmented vectors:
//   A row  = (-2qx, -2qy, |q|^2, 1)
//   B col  = ( rx,   ry,  1,  |r|^2)
// One 16x16x4 f32 WMMA produces a full 16x16 tile of exact fp32 squared
// distances with no epilogue arithmetic. A prologue kernel materializes the
// augmented B columns (float4 per point) in d_ws so the hot loop is just
// load -> wmma -> min/argmin update.

typedef __attribute__((ext_vector_type(2))) float v2f;
typedef __attribute__((ext_vector_type(8))) float v8f;

#define WAVES_PER_BLOCK 8
#define TILE 16

// aug[i] = (x, y, 1, x^2+y^2) for every point (flat over batches).
__global__ __launch_bounds__(256)
void augment_kernel(const float2* __restrict__ pts, float4* __restrict__ aug, int n) {
    int i = blockIdx.x * blockDim.x + threadIdx.x;
    if (i < n) {
        float2 p = pts[i];
        aug[i] = make_float4(p.x, p.y, 1.0f, p.x * p.x + p.y * p.y);
    }
}

template <bool PRE>
__global__ __launch_bounds__(WAVES_PER_BLOCK * 32)
void chamfer_pass_kernel(const float2* __restrict__ xq,   // [B, nq] query points (rows / A)
                         const void*   __restrict__ xr,   // PRE: float4 aug[B*nr]; else float2 raw
                         int nq, int nr,
                         float* __restrict__ dist,        // [B, nq] min_j d(q_i, r_j)
                         int*   __restrict__ idx)         // [B, nq] argmin_j (first occurrence)
{
    const int lane   = threadIdx.x & 31;
    const int wave   = threadIdx.x >> 5;
    const int wid    = blockIdx.x * WAVES_PER_BLOCK + wave;
    const int tilesQ = nq / TILE;
    const int b      = wid / tilesQ;
    const int i0     = (wid % tilesQ) * TILE;

    const int  lo = lane & 15;
    const bool hi = lane >= 16;   // lanes 16-31 hold the K=2,3 halves of A/B

    // ---- Build A (16x4 f32, 2 VGPRs), branchless so EXEC stays full ----
    // lanes 0-15 : (K0,K1) = (-2qx, -2qy)   for row M = lo
    // lanes 16-31: (K2,K3) = (|q|^2, 1.0f)  for row M = lo
    const float2* q  = xq + (size_t)b * nq;
    float2        qp = q[i0 + lo];
    v2f a;
    a.x = hi ? (qp.x * qp.x + qp.y * qp.y) : (-2.0f * qp.x);
    a.y = hi ? 1.0f                        : (-2.0f * qp.y);

    // B-side pointers.
    // PRE:  lane reads 8B of aug[j]: low lanes (.x,.y), high lanes (.z,.w).
    //       One wave-load covers 16 contiguous float4s (256B, fully coalesced).
    const float2* raug = (const float2*)((const float4*)xr + (size_t)b * nr) + (hi ? 1 : 0);
    const float2* rraw = (const float2*)xr + (size_t)b * nr;

    // Running per-lane min/argmin. D-layout: VGPR k <-> row M = k (+8 for hi
    // lanes), column N = lo. Lane's candidate column index is j0 + lo.
    float bestd[8];
    int   bestj[8];
#pragma unroll
    for (int k = 0; k < 8; ++k) { bestd[k] = 3.402823466e38f; bestj[k] = 0; }

#pragma unroll 4
    for (int j0 = 0; j0 < nr; j0 += TILE) {
        v2f bm;
        if (PRE) {
            float2 bv = raug[(size_t)(j0 + lo) * 2];
            bm.x = bv.x;
            bm.y = bv.y;
        } else {
            // lanes 0-15 : (K0,K1) = (rx, ry);  lanes 16-31: (K2,K3) = (1, |r|^2)
            float2 rp = rraw[j0 + lo];
            bm.x = hi ? 1.0f                        : rp.x;
            bm.y = hi ? (rp.x * rp.x + rp.y * rp.y) : rp.y;
        }

        v8f c = {0.f, 0.f, 0.f, 0.f, 0.f, 0.f, 0.f, 0.f};
        // D = A x B + 0 : full 16x16 tile of squared distances.
        v8f d = __builtin_amdgcn_wmma_f32_16x16x4_f32(
            /*neg_a=*/false, a, /*neg_b=*/false, bm,
            /*c_mod=*/(short)0, c, /*reuse_a=*/false, /*reuse_b=*/false);

        const int j = j0 + lo;
#pragma unroll
        for (int k = 0; k < 8; ++k) {
            float dv = d[k];
            bool  lt = dv < bestd[k];      // strict < keeps first occurrence
            bestd[k] = lt ? dv : bestd[k];
            bestj[k] = lt ? j  : bestj[k];
        }
    }

    // ---- Reduce across the 16 lanes of each half (columns of the tile). ----
    // Tie-break on smaller index -> exact jnp.argmin (first occurrence).
#pragma unroll
    for (int k = 0; k < 8; ++k) {
        float dv = bestd[k];
        int   jv = bestj[k];
#pragma unroll
        for (int m = 8; m >= 1; m >>= 1) {
            float od   = __shfl_xor(dv, m, 16);
            int   oj   = __shfl_xor(jv, m, 16);
            bool  take = (od < dv) || (od == dv && oj < jv);
            dv = take ? od : dv;
            jv = take ? oj : jv;
        }
        if (lo == 0) {
            int row = i0 + k + (hi ? 8 : 0);
            dist[(size_t)b * nq + row] = dv;
            idx [(size_t)b * nq + row] = jv;
        }
    }
}

extern "C" void kernel_launch(void* const* d_in, const int* in_sizes, int n_in,
                              void* d_out, int out_size, void* d_ws, size_t ws_size,
                              hipStream_t stream) {
    (void)in_sizes; (void)n_in; (void)out_size;

    const int B = 4, N = 8192, M = 8192;

    const float2* x1 = (const float2*)d_in[0];   // [B, N, 2]
    const float2* x2 = (const float2*)d_in[1];   // [B, M, 2]

    // d_out layout (return order, flat): dist1[B*N] f32, dist2[B*M] f32,
    // idx1[B*N] i32, idx2[B*M] i32.
    float* dist1 = (float*)d_out;
    float* dist2 = dist1 + (size_t)B * N;
    int*   idx1  = (int*)(dist2 + (size_t)B * M);
    int*   idx2  = idx1 + (size_t)B * N;

    const dim3 grid1((B * N / TILE) / WAVES_PER_BLOCK);
    const dim3 grid2((B * M / TILE) / WAVES_PER_BLOCK);
    const dim3 block(WAVES_PER_BLOCK * 32);

    const size_t need = (size_t)(B * N + B * M) * sizeof(float4);
    if (ws_size >= need) {
        float4* aug1 = (float4*)d_ws;            // B*N augmented set1 points
        float4* aug2 = aug1 + (size_t)B * N;     // B*M augmented set2 points
        augment_kernel<<<(B * N + 255) / 256, 256, 0, stream>>>(x1, aug1, B * N);
        augment_kernel<<<(B * M + 255) / 256, 256, 0, stream>>>(x2, aug2, B * M);
        chamfer_pass_kernel<true><<<grid1, block, 0, stream>>>(x1, aug2, N, M, dist1, idx1);
        chamfer_pass_kernel<true><<<grid2, block, 0, stream>>>(x2, aug1, M, N, dist2, idx2);
    } else {
        chamfer_pass_kernel<false><<<grid1, block, 0, stream>>>(x1, x2, N, M, dist1, idx1);
        chamfer_pass_kernel<false><<<grid2, block, 0, stream>>>(x2, x1, M, N, dist2, idx2);
    }
}